// DynamicBernoulliEmbeddingModel_54245436949088
// MI455X (gfx1250) — compile-verified
//
#include <hip/hip_runtime.h>

// ---- problem constants (match reference) ----
#define V_       50000
#define T_       10
#define K_       50
#define NS_      20
#define B_       16384
#define CTX_     8
#define EPS_     1e-7f
#define LAMBDA_  10000.0f
#define LAMBDA0_ 1.0f

#define NB_DRIFT 1024
#define NB_PRIOR 512
#define NB_MAIN  128   // 128 blocks * 8 waves * 16 items = 16384 = B_

typedef __attribute__((ext_vector_type(2))) float v2f;
typedef __attribute__((ext_vector_type(8))) float v8f;

// Deterministic block reduction (blockDim.x == 256, 8 wave32s).
// Result valid on thread 0.
__device__ __forceinline__ float blockReduce256(float v) {
    __shared__ float sm[8];
    #pragma unroll
    for (int off = 16; off > 0; off >>= 1)
        v += __shfl_down(v, off, 32);
    const int lane = threadIdx.x & 31;
    const int w    = threadIdx.x >> 5;
    if (lane == 0) sm[w] = v;
    __syncthreads();
    float r = 0.f;
    if (threadIdx.x == 0) {
        #pragma unroll
        for (int i = 0; i < 8; ++i) r += sm[i];
    }
    __syncthreads();   // protect sm for a subsequent call
    return r;
}

// ---- drift prior: sum over (rho_t[1:] - rho_t[:-1])^2, streams all of rho ----
// N = (T-1)*V*K = 22.5M floats; vectorized float4. Regular-temporal loads keep
// rho resident in the 192MB L2 for the gather phase that follows.
__global__ __launch_bounds__(256) void debm_drift(const float* __restrict__ rho,
                                                  float* __restrict__ part) {
    const long N4  = (long)(T_ - 1) * V_ * K_ / 4;  // 5,625,000
    const long VK4 = (long)V_ * K_ / 4;             // 625,000
    const float4* r4 = (const float4*)rho;
    float acc = 0.f;
    for (long i = (long)blockIdx.x * blockDim.x + threadIdx.x; i < N4;
         i += (long)gridDim.x * blockDim.x) {
        float4 a = r4[i];
        float4 b = r4[i + VK4];
        float dx = b.x - a.x, dy = b.y - a.y, dz = b.z - a.z, dw = b.w - a.w;
        acc += dx * dx + dy * dy + dz * dz + dw * dw;
    }
    float s = blockReduce256(acc);
    if (threadIdx.x == 0) part[blockIdx.x] = s;
}

// ---- embedding prior: sum(alpha^2) + sum(rho[0,:]^2) (first row, K elems) ----
__global__ __launch_bounds__(256) void debm_prior(const float* __restrict__ alpha,
                                                  const float* __restrict__ rho,
                                                  float* __restrict__ part) {
    const long N4 = (long)V_ * K_ / 4;  // 625,000
    const float4* a4 = (const float4*)alpha;
    float acc = 0.f;
    for (long i = (long)blockIdx.x * blockDim.x + threadIdx.x; i < N4;
         i += (long)gridDim.x * blockDim.x) {
        float4 a = a4[i];
        acc += a.x * a.x + a.y * a.y + a.z * a.z + a.w * a.w;
    }
    if (blockIdx.x == 0 && threadIdx.x < K_) {
        float r = rho[threadIdx.x];
        acc += r * r;
    }
    float s = blockReduce256(acc);
    if (threadIdx.x == 0) part[blockIdx.x] = s;
}

// ---- fused main: ctx-sum (masked gather) + 21 dots/item via WMMA + log terms ----
// One wave handles 16 batch items. Per candidate: D = A x B accumulated over 13
// K-chunks of 4 with V_WMMA_F32_16X16X4_F32, where
//   A[m][k] = ctx_sum[item m][kchunk+k]   (16x4)
//   B[k][n] = rho[cand(item n)][kchunk+k] (4x16)
// so diag(D)[i] = dot(ctx_sum[i], rho_cand[i]).
// A-layout (ISA 7.12.2): lane L<16 holds A[L][{0,1}] in v0/v1; lane L>=16 holds
// A[L-16][{2,3}]. B mirrors it transposed, so every lane feeds data of item
// (lane&15) at k-pair offset 2*(lane>>4) -- a single 8B load each for A and B.
// Diagonal (i,i): lane i comp i for i<8; lane i+16 comp i-8 for i>=8.
__global__ __launch_bounds__(256) void debm_main(const float* __restrict__ rho,
                                                 const float* __restrict__ alpha,
                                                 const int* __restrict__ targets,
                                                 const int* __restrict__ times,
                                                 const int* __restrict__ contexts,
                                                 const int* __restrict__ negs,
                                                 float* __restrict__ pos_part,
                                                 float* __restrict__ neg_part) {
    __shared__ float ctx[8][16][52];   // K padded to 52 with zeros (26.6 KB)
    const int tid  = threadIdx.x;
    const int lane = tid & 31;
    const int wave = tid >> 5;
    const int i    = lane & 15;        // item this lane feeds
    const int hi   = lane >> 4;        // 0: k-pair {0,1}, 1: k-pair {2,3}
    const int base = (blockIdx.x * 8 + wave) * 16;

    // ---- phase 1: masked context-embedding sums into LDS ----
    const int kp = 2 * lane;           // k-pair handled by this lane (lanes 0..24)
    for (int it = 0; it < 16; ++it) {
        const int bb = base + it;
        float a0 = 0.f, a1 = 0.f;
        #pragma unroll
        for (int c = 0; c < CTX_; ++c) {
            const int cx = contexts[bb * CTX_ + c];   // uniform across wave
            if (cx >= 0 && lane < 25) {
                const float* ap = alpha + (size_t)cx * K_ + kp;
                a0 += ap[0];
                a1 += ap[1];
            }
        }
        if (lane < 25) { ctx[wave][it][kp] = a0; ctx[wave][it][kp + 1] = a1; }
        if (lane == 25) { ctx[wave][it][50] = 0.f; ctx[wave][it][51] = 0.f; }
    }
    __syncthreads();

    // ---- phase 2: 1 positive + NS negatives per item via WMMA ----
    const int    b       = base + i;
    const size_t rowbase = (size_t)times[b] * V_;
    float accPos = 0.f, accNeg = 0.f;
    const bool own  = (lane < 8) || (lane >= 24);
    const int  comp = (lane < 8) ? lane : ((lane >= 24) ? (lane - 24) : 0);

    int cand = targets[b];
    for (int n = 0; n < 1 + NS_; ++n) {
        const float* rp = rho + (rowbase + (size_t)cand) * K_;
        if (n < NS_) {  // prefetch next candidate row (2 cachelines / row)
            const int cn = negs[b * NS_ + n];
            __builtin_prefetch(rho + (rowbase + (size_t)cn) * K_ + 32 * hi, 0, 3);
            cand = cn;
        }
        v8f acc = {};
        #pragma unroll
        for (int j = 0; j < 13; ++j) {
            const int kk = 4 * j + 2 * hi;
            const float a0 = ctx[wave][i][kk];       // 0 beyond K (padded)
            const float a1 = ctx[wave][i][kk + 1];
            float b0 = 0.f, b1 = 0.f;
            if (kk < K_) { b0 = rp[kk]; b1 = rp[kk + 1]; }
            v2f A  = { a0, a1 };
            v2f Bm = { b0, b1 };
            acc = __builtin_amdgcn_wmma_f32_16x16x4_f32(
                false, A, false, Bm, (short)0, acc, false, false);
        }
        // extract diag(D) on owning lanes
        float eta = 0.f;
        #pragma unroll
        for (int r = 0; r < 8; ++r) eta = (comp == r) ? acc[r] : eta;
        if (own) {
            if (n == 0) {  // stable log_sigmoid
                accPos += (eta >= 0.f) ? -log1pf(expf(-eta))
                                       : (eta - log1pf(expf(eta)));
            } else {       // log(1 - sigmoid(eta) + eps), as in reference
                const float sg = 1.f / (1.f + expf(-eta));
                accNeg += logf(1.f - sg + EPS_);
            }
        }
    }
    const float ps = blockReduce256(accPos);
    const float ns = blockReduce256(accNeg);
    if (tid == 0) { pos_part[blockIdx.x] = ps; neg_part[blockIdx.x] = ns; }
}

// ---- finalize: fixed-order reduction of partials, write 4 scalars ----
__global__ __launch_bounds__(256) void debm_final(const float* __restrict__ dp,
                                                  const float* __restrict__ pp,
                                                  const float* __restrict__ posp,
                                                  const float* __restrict__ negp,
                                                  float* __restrict__ out) {
    const int tid = threadIdx.x;
    float v = 0.f;
    for (int i = tid; i < NB_DRIFT; i += 256) v += dp[i];
    const float dsum = blockReduce256(v);
    v = 0.f;
    for (int i = tid; i < NB_PRIOR; i += 256) v += pp[i];
    const float psum = blockReduce256(v);
    v = (tid < NB_MAIN) ? posp[tid] : 0.f;
    const float possum = blockReduce256(v);
    v = (tid < NB_MAIN) ? negp[tid] : 0.f;
    const float negsum = blockReduce256(v);
    if (tid == 0) {
        const float L_pos   = possum;
        const float L_neg   = negsum;
        const float L_prior = -0.5f * LAMBDA0_ * psum - 0.5f * LAMBDA_ * dsum;
        out[0] = -(1.0f * (L_pos + L_neg) + L_prior);  // M = 1.0
        out[1] = L_pos;
        out[2] = L_neg;
        out[3] = L_prior;
    }
}

extern "C" void kernel_launch(void* const* d_in, const int* in_sizes, int n_in,
                              void* d_out, int out_size, void* d_ws, size_t ws_size,
                              hipStream_t stream) {
    const float* rho      = (const float*)d_in[0];
    const float* alpha    = (const float*)d_in[1];
    const int*   targets  = (const int*)d_in[2];
    const int*   times    = (const int*)d_in[3];
    const int*   contexts = (const int*)d_in[4];
    const int*   negs     = (const int*)d_in[5];
    float*       out      = (float*)d_out;

    float* ws         = (float*)d_ws;
    float* drift_part = ws;                       // NB_DRIFT
    float* prior_part = ws + NB_DRIFT;            // NB_PRIOR
    float* pos_part   = ws + NB_DRIFT + NB_PRIOR; // NB_MAIN
    float* neg_part   = pos_part + NB_MAIN;       // NB_MAIN

    // drift first: streams rho once and leaves it resident in the 192MB L2
    // so the gather-heavy main kernel hits L2 instead of HBM.
    debm_drift<<<NB_DRIFT, 256, 0, stream>>>(rho, drift_part);
    debm_prior<<<NB_PRIOR, 256, 0, stream>>>(alpha, rho, prior_part);
    debm_main<<<NB_MAIN, 256, 0, stream>>>(rho, alpha, targets, times,
                                           contexts, negs, pos_part, neg_part);
    debm_final<<<1, 256, 0, stream>>>(drift_part, prior_part, pos_part,
                                      neg_part, out);
}